// PointRend_try_49709951484602
// MI455X (gfx1250) — compile-verified
//
#include <hip/hip_runtime.h>

#define HC 256
#define WC 256
#define HW_C (HC * WC)     // 65536 coarse pixels per batch
#define CF 128
#define HF 512
#define WF 512
#define NB 4
#define NPTS 8192
#define HIDDEN 256
#define KPAD 132           // 129 features padded to multiple of 4

typedef __attribute__((ext_vector_type(2))) float v2f;
typedef __attribute__((ext_vector_type(8))) float v8f;

__device__ __forceinline__ unsigned absbits(float x) {
    return __float_as_uint(x) & 0x7fffffffu;   // monotonic key for |x| (no NaNs in inputs)
}

// ---------------------------------------------------------------------------
// 1) Per-batch radix select: threshold T = 8192-th smallest |logit| bits,
//    plus counters for the compaction pass. One workgroup per batch.
// ---------------------------------------------------------------------------
__global__ __launch_bounds__(256)
void pr_select_kernel(const float* __restrict__ coarse,
                      unsigned* __restrict__ T_out,
                      int* __restrict__ ctr_less,
                      int* __restrict__ ctr_eq) {
    const int b = blockIdx.x;
    const int tid = threadIdx.x;
    const float* src = coarse + (size_t)b * HW_C;

    __shared__ unsigned hist[256];
    __shared__ unsigned s_prefix;
    __shared__ unsigned s_mask;
    __shared__ int s_kth;

    if (tid == 0) { s_prefix = 0u; s_mask = 0u; s_kth = NPTS; }
    __syncthreads();

    for (int shift = 24; shift >= 0; shift -= 8) {
        hist[tid] = 0u;                  // 256 threads == 256 bins
        __syncthreads();
        const unsigned prefix = s_prefix;
        const unsigned mask = s_mask;
        for (int i = tid; i < HW_C; i += 256) {
            unsigned u = absbits(src[i]);
            if ((u & mask) == prefix)
                atomicAdd(&hist[(u >> shift) & 255u], 1u);
        }
        __syncthreads();
        if (tid == 0) {
            unsigned kth = (unsigned)s_kth;
            unsigned cum = 0, chosen = 255;
            for (unsigned d = 0; d < 256; ++d) {
                unsigned c = hist[d];
                if (cum + c >= kth) { chosen = d; break; }
                cum += c;
            }
            s_kth = (int)(kth - cum);            // still needed among digit==chosen
            s_prefix = prefix | (chosen << shift);
            s_mask = mask | (0xFFu << shift);
        }
        __syncthreads();
    }
    if (tid == 0) {
        T_out[b]    = s_prefix;          // full 32-bit threshold key
        ctr_less[b] = 0;                 // slots for keys strictly below T
        ctr_eq[b]   = NPTS - s_kth;      // base slot for keys equal to T
    }
}

// ---------------------------------------------------------------------------
// 2) Compaction: gather indices of the selected points into sel[b][0..8191].
//    Order within the array is irrelevant to the final output.
// ---------------------------------------------------------------------------
__global__ __launch_bounds__(256)
void pr_compact_kernel(const float* __restrict__ coarse,
                       const unsigned* __restrict__ T,
                       int* __restrict__ ctr_less,
                       int* __restrict__ ctr_eq,
                       int* __restrict__ sel) {
    const int gid = blockIdx.x * 256 + threadIdx.x;
    const int b = gid >> 16;
    const int i = gid & (HW_C - 1);
    const unsigned u = absbits(coarse[(size_t)b * HW_C + i]);
    const unsigned t = T[b];
    if (u < t) {
        int pos = atomicAdd(&ctr_less[b], 1);
        sel[b * NPTS + pos] = i;
    } else if (u == t) {
        int pos = atomicAdd(&ctr_eq[b], 1);
        if (pos < NPTS) sel[b * NPTS + pos] = i;   // take enough ties to fill k
    }
}

// ---------------------------------------------------------------------------
// 3) Gather + MLP (fp32 WMMA) + scatter. One workgroup = 16 points,
//    512 threads = 16 waves, wave w owns hidden columns [16w, 16w+16).
// ---------------------------------------------------------------------------
__global__ __launch_bounds__(512)
void pr_mlp_kernel(const float* __restrict__ coarse,
                   const float* __restrict__ fine,
                   const float* __restrict__ W1,
                   const float* __restrict__ b1,
                   const float* __restrict__ W2,
                   const float* __restrict__ b2,
                   const int* __restrict__ sel,
                   float* __restrict__ out) {
    const int b    = blockIdx.x >> 9;    // 512 tiles per batch
    const int tile = blockIdx.x & 511;
    const int t    = threadIdx.x;
    const int lane = t & 31;
    const int w    = t >> 5;             // wave id 0..15

    __shared__ float feats[16][KPAD];    // A tiles, row-major, K padded with zeros
    __shared__ int   s_idx[16];
    __shared__ float wave_part[16][16];  // [wave][point-row]

    if (t < 16) {
        int idx = sel[b * NPTS + tile * 16 + t];
        s_idx[t] = idx;
        feats[t][128] = coarse[(size_t)b * HW_C + idx];  // appended coarse logit
        feats[t][129] = 0.f; feats[t][130] = 0.f; feats[t][131] = 0.f;
    }
    __syncthreads();

    // Bilinear gather. With H_F = 2*H_C the sample point is (2*ix+0.5, 2*iy+0.5):
    // weights are all 0.25 and always in-bounds -> plain 2x2 average.
    {
        const int p  = t >> 5;
        const int c0 = t & 31;
        const int idx = s_idx[p];
        const int x0 = (idx & (WC - 1)) * 2;
        const int y0 = (idx >> 8) * 2;
        const float* fb = fine + (size_t)b * CF * HF * WF + (size_t)y0 * WF + x0;
        for (int c = c0; c < CF; c += 32) {
            const float* fp = fb + (size_t)c * (HF * WF);
            v2f r0 = *(const v2f*)(fp);        // (y0, x0..x0+1), 8B aligned (x0 even)
            v2f r1 = *(const v2f*)(fp + WF);   // (y0+1, ...)
            feats[p][c] = ((r0.x * 0.25f + r0.y * 0.25f) + r1.x * 0.25f) + r1.y * 0.25f;
        }
    }
    __syncthreads();

    // K-loop of V_WMMA_F32_16X16X4_F32.
    // fp32 A 16x4 layout: lanes 0-15 rows, VGPR0 = K0 (lo half) / K2 (hi half),
    // VGPR1 = K1 / K3. B 4x16 mirrored: VGPR0 = K0/K2 rows, VGPR1 = K1/K3.
    const int n      = (w << 4) + (lane & 15);  // global hidden index
    const int row    = lane & 15;
    const int kdelta = (lane < 16) ? 0 : 2;

    v8f acc = {};
    for (int kb = 0; kb < KPAD / 4; ++kb) {
        const int k0 = kb * 4;
        const int kr = k0 + kdelta;
        v2f a = *(const v2f*)&feats[row][kr];   // 8B aligned (row stride 528, kr even)
        v2f bb;
        bb.x = (kr < 129)     ? W1[kr * HIDDEN + n]       : 0.f;
        bb.y = (kr + 1 < 129) ? W1[(kr + 1) * HIDDEN + n] : 0.f;
        acc = __builtin_amdgcn_wmma_f32_16x16x4_f32(false, a, false, bb,
                                                    (short)0, acc, false, false);
    }

    // Bias + ReLU + W2, then deterministic reduction over the 16 N-lanes
    // (butterfly within each 16-lane half; masks 1,2,4,8 stay inside the half).
    const float b1n = b1[n];
    const float w2n = W2[n];
    const int base_row = (lane < 16) ? 0 : 8;   // C/D layout: hi half holds M+8
    #pragma unroll
    for (int i = 0; i < 8; ++i) {
        float h = acc[i] + b1n;
        h = fmaxf(h, 0.f);
        float pv = h * w2n;
        pv += __shfl_xor(pv, 1, 32);
        pv += __shfl_xor(pv, 2, 32);
        pv += __shfl_xor(pv, 4, 32);
        pv += __shfl_xor(pv, 8, 32);
        if ((lane & 15) == 0) wave_part[w][base_row + i] = pv;
    }
    __syncthreads();

    // Fixed-order cross-wave sum -> deterministic fp32 result; scatter.
    if (t < 16) {
        float s = 0.f;
        #pragma unroll
        for (int ww = 0; ww < 16; ++ww) s += wave_part[ww][t];
        out[(size_t)b * HW_C + s_idx[t]] = s + b2[0];
    }
}

// ---------------------------------------------------------------------------
extern "C" void kernel_launch(void* const* d_in, const int* in_sizes, int n_in,
                              void* d_out, int out_size, void* d_ws, size_t ws_size,
                              hipStream_t stream) {
    const float* coarse = (const float*)d_in[0];
    const float* fine   = (const float*)d_in[1];
    const float* W1     = (const float*)d_in[2];
    const float* b1     = (const float*)d_in[3];
    const float* W2     = (const float*)d_in[4];
    const float* b2     = (const float*)d_in[5];
    float* out = (float*)d_out;

    char* ws = (char*)d_ws;
    unsigned* T   = (unsigned*)ws;           // 4 x u32
    int* ctr_less = (int*)(ws + 16);         // 4 x i32
    int* ctr_eq   = (int*)(ws + 32);         // 4 x i32
    int* sel      = (int*)(ws + 64);         // 4 x 8192 x i32

    // 1) out = coarse_logits (refined points overwritten below)
    hipMemcpyAsync(out, coarse, (size_t)NB * HW_C * sizeof(float),
                   hipMemcpyDeviceToDevice, stream);
    // 2) per-batch threshold for the 8192 smallest |logit|
    pr_select_kernel<<<NB, 256, 0, stream>>>(coarse, T, ctr_less, ctr_eq);
    // 3) gather selected indices
    pr_compact_kernel<<<(NB * HW_C) / 256, 256, 0, stream>>>(coarse, T, ctr_less, ctr_eq, sel);
    // 4) feature gather + WMMA MLP + scatter
    pr_mlp_kernel<<<NB * (NPTS / 16), 512, 0, stream>>>(coarse, fine, W1, b1, W2, b2, sel, out);
}